// MyModel_87454124081981
// MI455X (gfx1250) — compile-verified
//
#include <hip/hip_runtime.h>
#include <hip/hip_bf16.h>
#include <math.h>

// CDNA5 / gfx1250 implementation.
// Precision: tolerances are 1e-6 -> all matrix math uses V_WMMA_F32_16X16X4_F32
// (fp32 in, fp32 accumulate). Depthwise conv is recast as 15 GEMMs
// (M=4096 batch, K=1024 in-pixels, N=1024 out-pixels) with B-fragments
// synthesized branchlessly from the 27x27x15 filter table in LDS
// (out-of-window taps read from a zeroed LDS pad via index-select, so the
// inner loop has NO exec-mask toggling around the WMMAs).

typedef __attribute__((ext_vector_type(2))) float v2f;
typedef __attribute__((ext_vector_type(8))) float v8f;

__device__ __forceinline__ v8f wmma_f32x4(v2f a, v2f b, v8f c) {
  // 8 args: (neg_a, A, neg_b, B, c_mod, C, reuse_a, reuse_b)
  return __builtin_amdgcn_wmma_f32_16x16x4_f32(false, a, false, b, (short)0, c,
                                               false, false);
}

// ---------------------------------------------------------------------------
// Kernel 1: TimeDistributed LSTM. One thread = one of 131072 sequences.
// Writes h_t (4 floats) into td_field laid out as (B, r*128 + t*4 + k).
// ---------------------------------------------------------------------------
__global__ __launch_bounds__(256) void lstm_kernel(
    const float* __restrict__ x, const float* __restrict__ W,
    const float* __restrict__ U, float* __restrict__ td_field) {
  __shared__ float sW[48];  // (3,16)
  __shared__ float sU[64];  // (4,16)
  int tid = threadIdx.x;
  for (int i = tid; i < 48; i += 256) sW[i] = W[i];
  for (int i = tid; i < 64; i += 256) sU[i] = U[i];
  __syncthreads();

  int n = blockIdx.x * 256 + tid;  // sequence id in [0, 131072)
  int b = n >> 5;
  int r = n & 31;
  const float* xb = x + (size_t)b * 3072 + (size_t)r * 96;  // (t,f), f fastest
  float* outb = td_field + (size_t)b * 4096 + (size_t)r * 128;

  float h[4] = {0.f, 0.f, 0.f, 0.f};
  float c[4] = {0.f, 0.f, 0.f, 0.f};

  for (int t = 0; t < 32; ++t) {
    float x0 = xb[t * 3 + 0], x1 = xb[t * 3 + 1], x2 = xb[t * 3 + 2];
    float z[16];
#pragma unroll
    for (int g = 0; g < 16; ++g) {
      float a = x0 * sW[g] + x1 * sW[16 + g] + x2 * sW[32 + g];
      a += h[0] * sU[g] + h[1] * sU[16 + g] + h[2] * sU[32 + g] +
           h[3] * sU[48 + g];
      z[g] = a;
    }
    float iv[4], fv[4], ov[4];
#pragma unroll
    for (int k = 0; k < 4; ++k) {
      float zi = z[k], zf = z[4 + k], zo = z[12 + k];
      iv[k] = zi > 0.f ? zi : expm1f(zi);
      fv[k] = zf > 0.f ? zf : expm1f(zf);
      ov[k] = zo > 0.f ? zo : expm1f(zo);
    }
    // softmax over zc
    float m = fmaxf(fmaxf(z[8], z[9]), fmaxf(z[10], z[11]));
    float e[4], s = 0.f;
#pragma unroll
    for (int k = 0; k < 4; ++k) { e[k] = expf(z[8 + k] - m); s += e[k]; }
    float inv = 1.f / s;
#pragma unroll
    for (int k = 0; k < 4; ++k) c[k] = fv[k] * c[k] + iv[k] * (e[k] * inv);
    // softmax over c
    float m2 = fmaxf(fmaxf(c[0], c[1]), fmaxf(c[2], c[3]));
    float e2[4], s2 = 0.f;
#pragma unroll
    for (int k = 0; k < 4; ++k) { e2[k] = expf(c[k] - m2); s2 += e2[k]; }
    float inv2 = 1.f / s2;
#pragma unroll
    for (int k = 0; k < 4; ++k) {
      h[k] = ov[k] * (e2[k] * inv2);
      outb[t * 4 + k] = h[k];
    }
  }
}

// ---------------------------------------------------------------------------
// Kernel 2: depthwise 27x27 (SAME) + pointwise + sigmoid, as 15 WMMA-f32 GEMMs.
// One wave per 16(batch) x 16(pixel) output tile; 15 fp32 accumulator tiles.
// LDS filter table has a zeroed pad slot so out-of-window taps are handled by
// index selection (v_cndmask on the address), never by exec-masked loads.
// ---------------------------------------------------------------------------
#define DW_PAD 10935               // start of zero pad in sdw
__global__ __launch_bounds__(32) void sepconv_kernel(
    const float* __restrict__ x, const float* __restrict__ dw,
    const float* __restrict__ pw, const float* __restrict__ bias,
    float* __restrict__ sc_field) {
  __shared__ float sdw[10960];     // 27*27*15 = 10935 floats + zero pad
  int lane = threadIdx.x;
  for (int i = lane; i < 10935; i += 32) sdw[i] = dw[i];
  for (int i = DW_PAD + lane; i < 10960; i += 32) sdw[i] = 0.f;
  __syncthreads();

  int b0 = blockIdx.x * 16;  // batch tile (M)
  int p0 = blockIdx.y * 16;  // output-pixel tile (N)
  int hi = lane >> 4;        // which half of the wave
  int lo = lane & 15;

  int p = p0 + lo;           // this lane's N (output pixel)
  int py = p >> 5, px = p & 31;
  const float* arow = x + (size_t)(b0 + lo) * 3072;  // this lane's M row

  v8f acc[15];
#pragma unroll
  for (int i = 0; i < 15; ++i)
#pragma unroll
    for (int j = 0; j < 8; ++j) acc[i][j] = 0.f;

#pragma unroll 2
  for (int q0 = 0; q0 < 1024; q0 += 4) {
    int k0 = q0 + 2 * hi;  // K index held in vgpr slot 0 of this lane
    // prefetch the A stream ~12 K-steps ahead (144 floats along the row)
    __builtin_prefetch(arow + (k0 + 48) * 3, 0, 3);
    int base[2];
#pragma unroll
    for (int v = 0; v < 2; ++v) {
      int q = k0 + v;                 // input pixel (K dim)
      int qy = q >> 5, qx = q & 31;
      int u = qy - py + 13;           // filter tap row
      int w = qx - px + 13;           // filter tap col
      bool ok = (u >= 0) && (u < 27) && (w >= 0) && (w < 27);
      int t = (u * 27 + w) * 15;
      base[v] = ok ? t : DW_PAD;      // branchless: pad region reads 0.0f
    }
    // A fragments: one per input channel, shared across 5 multipliers.
    v2f A[3];
#pragma unroll
    for (int c = 0; c < 3; ++c) {
      A[c].x = arow[(k0 + 0) * 3 + c];
      A[c].y = arow[(k0 + 1) * 3 + c];
    }
#pragma unroll
    for (int cm = 0; cm < 15; ++cm) {
      v2f B;
      B.x = sdw[base[0] + cm];        // unconditional ds_load_b32
      B.y = sdw[base[1] + cm];
      acc[cm] = wmma_f32x4(A[cm / 5], B, acc[cm]);
    }
  }

  // Fused pointwise (15 -> 4) + bias + sigmoid epilogue.
  float pwr[15][4];
#pragma unroll
  for (int cm = 0; cm < 15; ++cm)
#pragma unroll
    for (int f = 0; f < 4; ++f) pwr[cm][f] = pw[cm * 4 + f];
  float bz[4];
#pragma unroll
  for (int f = 0; f < 4; ++f) bz[f] = bias[f];

#pragma unroll
  for (int g = 0; g < 8; ++g) {
    int row = b0 + g + 8 * hi;  // C/D layout: vgpr g -> M = g + 8*(lane/16)
    float o4[4] = {bz[0], bz[1], bz[2], bz[3]};
#pragma unroll
    for (int cm = 0; cm < 15; ++cm) {
      float d = acc[cm][g];
#pragma unroll
      for (int f = 0; f < 4; ++f) o4[f] += d * pwr[cm][f];
    }
    float4 s;
    s.x = 1.f / (1.f + expf(-o4[0]));
    s.y = 1.f / (1.f + expf(-o4[1]));
    s.z = 1.f / (1.f + expf(-o4[2]));
    s.w = 1.f / (1.f + expf(-o4[3]));
    *(float4*)(sc_field + (size_t)row * 4096 + (size_t)p * 4) = s;
  }
}

// ---------------------------------------------------------------------------
// Kernel 3: (4096,4096) @ (4096,10) via WMMA f32 (N padded to 16).
// One wave per 16-row tile. Column masking is branchless: the B index is
// clamped in-bounds (Bw has 40960 elements) and the value zeroed by select.
// ---------------------------------------------------------------------------
__global__ __launch_bounds__(32) void field_gemm_kernel(
    const float* __restrict__ A, const float* __restrict__ Bw,
    float* __restrict__ out) {
  int lane = threadIdx.x;
  int hi = lane >> 4;
  int lo = lane & 15;
  int m0 = blockIdx.x * 16;
  const float* arow = A + (size_t)(m0 + lo) * 4096;

  v8f acc;
#pragma unroll
  for (int j = 0; j < 8; ++j) acc[j] = 0.f;

  bool col_ok = lo < 10;
  int col = col_ok ? lo : 9;  // clamped, always in-bounds
#pragma unroll 4
  for (int k0 = 0; k0 < 4096; k0 += 4) {
    int k = k0 + 2 * hi;
    __builtin_prefetch(arow + k + 256, 0, 3);  // stream A 1 KB ahead
    v2f a, b;
    a.x = arow[k];
    a.y = arow[k + 1];
    float b0v = Bw[(size_t)k * 10 + col];        // unconditional load
    float b1v = Bw[(size_t)(k + 1) * 10 + col];
    b.x = col_ok ? b0v : 0.f;                    // v_cndmask, no branch
    b.y = col_ok ? b1v : 0.f;
    acc = wmma_f32x4(a, b, acc);
  }

  if (col_ok) {
#pragma unroll
    for (int g = 0; g < 8; ++g)
      out[(size_t)(m0 + g + 8 * hi) * 10 + lo] = acc[g];
  }
}

// ---------------------------------------------------------------------------
// Kernel 4: tolerance compare -> 1.0f / 0.0f
// ---------------------------------------------------------------------------
__global__ void compare_kernel(const float* __restrict__ td,
                               const float* __restrict__ sc,
                               float* __restrict__ out, int n) {
  int i = blockIdx.x * blockDim.x + threadIdx.x;
  if (i < n) {
    float a = td[i], b = sc[i];
    float tol = 1e-6f + 1e-6f * fabsf(b);
    out[i] = (fabsf(a - b) <= tol) ? 1.0f : 0.0f;
  }
}

extern "C" void kernel_launch(void* const* d_in, const int* in_sizes, int n_in,
                              void* d_out, int out_size, void* d_ws,
                              size_t ws_size, hipStream_t stream) {
  (void)in_sizes; (void)n_in; (void)out_size; (void)ws_size;
  const float* x     = (const float*)d_in[0];  // (4096,32,32,3)
  const float* Wl    = (const float*)d_in[1];  // (3,16)
  const float* Ul    = (const float*)d_in[2];  // (4,16)
  const float* dwk   = (const float*)d_in[3];  // (27,27,3,5)
  const float* pwk   = (const float*)d_in[4];  // (15,4)
  const float* cbias = (const float*)d_in[5];  // (4,)
  const float* W_td  = (const float*)d_in[6];  // (4096,10)
  const float* W_sc  = (const float*)d_in[7];  // (4096,10)

  float* ws = (float*)d_ws;
  const size_t FIELD = 4096ull * 4096ull;   // 16.78M floats each
  float* td_field = ws;
  float* sc_field = ws + FIELD;
  float* td_out   = ws + 2 * FIELD;          // 4096*10
  float* sc_out   = td_out + 40960;
  float* out      = (float*)d_out;

  hipLaunchKernelGGL(lstm_kernel, dim3(512), dim3(256), 0, stream,
                     x, Wl, Ul, td_field);
  hipLaunchKernelGGL(sepconv_kernel, dim3(256, 64), dim3(32), 0, stream,
                     x, dwk, pwk, cbias, sc_field);
  hipLaunchKernelGGL(field_gemm_kernel, dim3(256), dim3(32), 0, stream,
                     td_field, W_td, td_out);
  hipLaunchKernelGGL(field_gemm_kernel, dim3(256), dim3(32), 0, stream,
                     sc_field, W_sc, sc_out);
  hipLaunchKernelGGL(compare_kernel, dim3((40960 + 255) / 256), dim3(256), 0,
                     stream, td_out, sc_out, out, 40960);
}